// HyperpriorDensity_12618613916309
// MI455X (gfx1250) — compile-verified
//
#include <hip/hip_runtime.h>
#include <hip/hip_bf16.h>
#include <stdint.h>

#define NCH   320
#define HW    1024      // 32*32
#define PSTRIDE 64      // floats per channel in packed workspace (59 used)

typedef unsigned int u32;
typedef u32 v4u __attribute__((ext_vector_type(4)));
typedef int  v8i __attribute__((ext_vector_type(8)));
typedef int  v4i __attribute__((ext_vector_type(4)));

__device__ __forceinline__ float fast_tanh(float x) {
#if __has_builtin(__builtin_amdgcn_tanhf)
    return __builtin_amdgcn_tanhf(x);        // v_tanh_f32 (confirmed taken)
#else
    float r;
    asm("v_tanh_f32 %0, %1\n\tv_nop" : "=v"(r) : "v"(x));
    return r;
#endif
}

__device__ __forceinline__ float fast_rcp(float x) {
#if __has_builtin(__builtin_amdgcn_rcpf)
    return __builtin_amdgcn_rcpf(x);         // v_rcp_f32
#else
    return 1.0f / x;
#endif
}

__device__ __forceinline__ float softplusf(float x) {
    // max(x,0) + log(1+exp(-|x|)): v_exp_f32 + v_log_f32 fast path.
    // (abs err ~1e-8 vs log1p in the deep tail; weights are O(1), harmless)
    return fmaxf(x, 0.0f) + __logf(1.0f + __expf(-fabsf(x)));
}

__device__ __forceinline__ float sigmoidf_fast(float z) {
    return fast_rcp(1.0f + __expf(-z));      // v_exp_f32 + v_rcp_f32
}

// ---------------------------------------------------------------------------
// MLP cdf-logit for one scalar, all constants in registers.
// activation: h = pre + t * tanh(pre)
// ---------------------------------------------------------------------------
__device__ __forceinline__ float mlp_cdf(float v, const float* P) {
    float h0 = fmaf(P[0], v, P[3]); h0 = fmaf(P[6], fast_tanh(h0), h0);
    float h1 = fmaf(P[1], v, P[4]); h1 = fmaf(P[7], fast_tanh(h1), h1);
    float h2 = fmaf(P[2], v, P[5]); h2 = fmaf(P[8], fast_tanh(h2), h2);
#pragma unroll
    for (int L = 0; L < 3; ++L) {
        const float* Q = P + 9 + 15 * L;
        float p0 = fmaf(Q[0], h0, fmaf(Q[1], h1, fmaf(Q[2], h2, Q[9])));
        float p1 = fmaf(Q[3], h0, fmaf(Q[4], h1, fmaf(Q[5], h2, Q[10])));
        float p2 = fmaf(Q[6], h0, fmaf(Q[7], h1, fmaf(Q[8], h2, Q[11])));
        h0 = fmaf(Q[12], fast_tanh(p0), p0);
        h1 = fmaf(Q[13], fast_tanh(p1), p1);
        h2 = fmaf(Q[14], fast_tanh(p2), p2);
    }
    float p = fmaf(P[54], h0, fmaf(P[55], h1, fmaf(P[56], h2, P[57])));
    return fmaf(P[58], fast_tanh(p), p);
}

// ---------------------------------------------------------------------------
// Main: one block per (channel, batch-image) contiguous 1024-float chunk.
// Per-channel constants DMA'd into LDS with the Tensor Data Mover.
// ---------------------------------------------------------------------------
__global__ void __launch_bounds__(256) hp_main(const float* __restrict__ x,
                                               const float* __restrict__ ws,
                                               float* __restrict__ out) {
    const int c = blockIdx.x;
    const int n = blockIdx.y;

    __shared__ float sw[PSTRIDE];

#if __has_builtin(__builtin_amdgcn_tensor_load_to_lds)
    if (threadIdx.x < 32) {  // wave 0 issues the TDM descriptor (EXEC ignored by TDM)
        unsigned long long gaddr =
            (unsigned long long)(uintptr_t)(ws + (size_t)c * PSTRIDE);
        u32 lds = (u32)(uintptr_t)(&sw[0]);  // flat shared addr: low 32 bits = LDS offset

        // D# group 0: count=1 | lds_addr | global_addr[56:0] | type=2
        v4u g0 = { 1u,
                   lds,
                   (u32)(gaddr & 0xFFFFFFFFu),
                   (u32)((gaddr >> 32) & 0x1FFFFFFu) | 0x80000000u };
        // D# group 1: data_size=4B, tensor_dim0=64, tensor_dim1=1,
        //             tile_dim0=64, tile_dim1=1, dim0_stride=64
        v8i g1 = { (int)0x00020000,        // wg_mask=0, data_size=2 (4B)
                   (int)(PSTRIDE << 16),   // tensor_dim0 lo16 @ [31:16]
                   (int)(1 << 16),         // tensor_dim0 hi=0; tensor_dim1 lo16=1
                   (int)(PSTRIDE << 16),   // tensor_dim1 hi=0; tile_dim0=64
                   (int)1,                 // tile_dim1=1; tile_dim2=0 (unused)
                   (int)PSTRIDE,           // tensor_dim0_stride lo32
                   0, 0 };
        v4i gz4 = {0, 0, 0, 0};
        v8i gz8 = {0, 0, 0, 0, 0, 0, 0, 0};
        __builtin_amdgcn_tensor_load_to_lds(g0, g1, gz4, gz4, gz8, 0);
        __builtin_amdgcn_s_wait_tensorcnt(0);
    }
#else
    if (threadIdx.x < PSTRIDE) sw[threadIdx.x] = ws[(size_t)c * PSTRIDE + threadIdx.x];
#endif
    __syncthreads();

    // hoist constants LDS -> registers (59 VGPRs, reused 8x per thread)
    float P[59];
#pragma unroll
    for (int i = 0; i < 59; ++i) P[i] = sw[i];

    // x/out layout (N,C,H,W): element m of channel c lives at n*C*HW + c*HW + hw
    const size_t base = ((size_t)n * NCH + c) * HW + (size_t)threadIdx.x * 4;
    const float4 v = *(const float4*)(x + base);
    float in[4] = {v.x, v.y, v.z, v.w};
    float r[4];
#pragma unroll
    for (int e = 0; e < 4; ++e) {
        float cu = mlp_cdf(in[e] + 0.5f, P);
        float cl = mlp_cdf(in[e] - 0.5f, P);
        float s  = cu + cl;
        float sg = (s < 0.0f) ? 1.0f : ((s > 0.0f) ? -1.0f : 0.0f);  // -sign(cu+cl)
        float lik = fabsf(sigmoidf_fast(sg * cu) - sigmoidf_fast(sg * cl));
        r[e] = fmaxf(lik, 1e-9f);
    }
    float4 o = {r[0], r[1], r[2], r[3]};
    *(float4*)(out + base) = o;
}

// ---------------------------------------------------------------------------
// Prep: fold softplus(H) and tanh(a) into packed per-channel constants.
// Layout per channel (stride 64 floats):
//   [0..2]  w0      [3..5]  b0      [6..8]  t0
//   [9+15L .. ] for L=0..2:  w[9] b[3] t[3]   (layers 1..3)
//   [54..56] w4     [57] b4         [58] t4
// ---------------------------------------------------------------------------
__global__ void hp_prep(const float* __restrict__ H0, const float* __restrict__ a0, const float* __restrict__ b0,
                        const float* __restrict__ H1, const float* __restrict__ a1, const float* __restrict__ b1,
                        const float* __restrict__ H2, const float* __restrict__ a2, const float* __restrict__ b2,
                        const float* __restrict__ H3, const float* __restrict__ a3, const float* __restrict__ b3,
                        const float* __restrict__ H4, const float* __restrict__ a4, const float* __restrict__ b4,
                        float* __restrict__ ws) {
    int c = blockIdx.x * blockDim.x + threadIdx.x;
    if (c >= NCH) return;
    float* P = ws + (size_t)c * PSTRIDE;

#pragma unroll
    for (int i = 0; i < 3; ++i) {
        P[i]     = softplusf(H0[c * 3 + i]);   // (C,3,1)
        P[3 + i] = b0[c * 3 + i];
        P[6 + i] = fast_tanh(a0[c * 3 + i]);
    }
    const float* Hs[3] = {H1, H2, H3};
    const float* As[3] = {a1, a2, a3};
    const float* Bs[3] = {b1, b2, b3};
#pragma unroll
    for (int L = 0; L < 3; ++L) {
        float* Q = P + 9 + 15 * L;
#pragma unroll
        for (int ij = 0; ij < 9; ++ij) Q[ij] = softplusf(Hs[L][c * 9 + ij]);   // (C,3,3) row-major i*3+j
#pragma unroll
        for (int i = 0; i < 3; ++i) {
            Q[9 + i]  = Bs[L][c * 3 + i];
            Q[12 + i] = fast_tanh(As[L][c * 3 + i]);
        }
    }
#pragma unroll
    for (int j = 0; j < 3; ++j) P[54 + j] = softplusf(H4[c * 3 + j]);          // (C,1,3)
    P[57] = b4[c];
    P[58] = fast_tanh(a4[c]);
#pragma unroll
    for (int i = 59; i < PSTRIDE; ++i) P[i] = 0.0f;
}

extern "C" void kernel_launch(void* const* d_in, const int* in_sizes, int n_in,
                              void* d_out, int out_size, void* d_ws, size_t ws_size,
                              hipStream_t stream) {
    const float* x  = (const float*)d_in[0];
    const float* H0 = (const float*)d_in[1];
    const float* a0 = (const float*)d_in[2];
    const float* b0 = (const float*)d_in[3];
    const float* H1 = (const float*)d_in[4];
    const float* a1 = (const float*)d_in[5];
    const float* b1 = (const float*)d_in[6];
    const float* H2 = (const float*)d_in[7];
    const float* a2 = (const float*)d_in[8];
    const float* b2 = (const float*)d_in[9];
    const float* H3 = (const float*)d_in[10];
    const float* a3 = (const float*)d_in[11];
    const float* b3 = (const float*)d_in[12];
    const float* H4 = (const float*)d_in[13];
    const float* a4 = (const float*)d_in[14];
    const float* b4 = (const float*)d_in[15];

    float* ws  = (float*)d_ws;   // needs 320*64*4 = 80 KB
    float* out = (float*)d_out;

    const int N = in_sizes[0] / (NCH * HW);   // 32

    hp_prep<<<(NCH + 63) / 64, 64, 0, stream>>>(H0, a0, b0, H1, a1, b1, H2, a2, b2,
                                                H3, a3, b3, H4, a4, b4, ws);

    dim3 grid(NCH, N);
    hp_main<<<grid, 256, 0, stream>>>(x, ws, out);
}